// MetaSpace_68822555951687
// MI455X (gfx1250) — compile-verified
//
#include <hip/hip_runtime.h>
#include <hip/hip_bf16.h>

typedef float v2f __attribute__((ext_vector_type(2)));
typedef float v8f __attribute__((ext_vector_type(8)));

#define BATCH 32
#define NKPT  17
#define NHEADS 8

// ---------------------------------------------------------------------------
// Wave32 all-reduce (butterfly): every lane ends with the sum.
// ---------------------------------------------------------------------------
__device__ __forceinline__ float wave_allreduce(float v) {
#pragma unroll
    for (int off = 16; off > 0; off >>= 1)
        v += __shfl_xor(v, off, 32);
    return v;
}

// ---------------------------------------------------------------------------
// 1) Gaussian keypoint pooling: kf[b*N+n, c] = sum_{5x5} g(dy,dx)*fm[b,c,iy,ix]
//    Separable gaussian, sigma=2, ksz=5; rintf matches jnp.round (half-even).
// ---------------------------------------------------------------------------
__global__ __launch_bounds__(256) void gauss_pool_kernel(
    const float* __restrict__ fm, const float* __restrict__ kpts,
    float* __restrict__ kf, int C, int H, int W) {
    int bn = blockIdx.x;                 // b*NKPT + n
    float kx = kpts[bn * 2 + 0] * ((float)W / 192.0f);
    float ky = kpts[bn * 2 + 1] * ((float)H / 256.0f);
    int cx = (int)rintf(kx);
    int cy = (int)rintf(ky);
    int ix[5], iy[5];
    float w1d[5];
    float s = 0.0f;
#pragma unroll
    for (int i = 0; i < 5; ++i) {
        int x = cx + i - 2;
        int y = cy + i - 2;
        ix[i] = min(max(x, 0), W - 1);
        iy[i] = min(max(y, 0), H - 1);
        float dof = (float)(i - 2);
        w1d[i] = expf(-(dof * dof) / 8.0f);
        s += w1d[i];
    }
    float inv = 1.0f / (s * s);
    int b = bn / NKPT;
    for (int c = threadIdx.x; c < C; c += blockDim.x) {
        const float* base = fm + (size_t)(b * C + c) * (size_t)(H * W);
        float acc = 0.0f;
#pragma unroll
        for (int yy = 0; yy < 5; ++yy) {
            const float* rowp = base + iy[yy] * W;
            float wy = w1d[yy];
#pragma unroll
            for (int xx = 0; xx < 5; ++xx)
                acc += wy * w1d[xx] * rowp[ix[xx]];
        }
        kf[(size_t)bn * C + c] = acc * inv;
    }
}

// ---------------------------------------------------------------------------
// 2) WMMA f32 GEMM:  D = Aop(M,K) x Bm(K,Nc) + bias[col] (+ extra[M,Nc])
//    AMODE 0: A plain row-major (lda = K)
//    AMODE 1: A(m,k) = k<C ? kf[m,k] : ms[m%NKPT, k-C]           (K = 2C)
//    AMODE 2: A(r,k) = (r even) ? kf[r/2,k] : ms[(r/2)%NKPT, k]  (K = C)
//    One wave computes a 16x64 tile (4 accumulators, A fragment reused 4x);
//    2 waves per block cover 32 rows.  Inner loop is software-pipelined:
//    iteration i+1's A (b64) + 4x B fragment loads are issued into fresh
//    registers before iteration i's 4 WMMAs, so the loads overlap the matrix
//    pipe instead of draining loadcnt to 0 at each WMMA.  Addresses are
//    strength-reduced pointer increments (no per-iteration 64-bit multiply).
//    All dims divide exactly (M%32==0, Nc%64==0, K%4==0) for every GEMM here.
// ---------------------------------------------------------------------------
__device__ __forceinline__ void gemm_inner(
    const float* __restrict__ arow,   // lane's A row (element k at [k-kAoff])
    const float* __restrict__ Bm, int Nc, int n0, int r, int half,
    int kbeg, int kend, int kAoff, v8f acc[4]) {
    int iters = (kend - kbeg) >> 2;                    // >= 1 always
    const float* aptr = arow + (kbeg - kAoff) + half * 2;
    const float* bptr = Bm + (size_t)(kbeg + half * 2) * Nc + n0 + r;
    const size_t bstep = (size_t)Nc * 4;

    // prologue: fragments for iteration 0
    v2f afr = *(const v2f*)aptr;
    v2f bf[4];
#pragma unroll
    for (int t = 0; t < 4; ++t) {
        bf[t].x = bptr[t * 16];
        bf[t].y = bptr[t * 16 + Nc];
    }

    for (int i = 1; i < iters; ++i) {
        aptr += 4;
        bptr += bstep;
        // issue next iteration's loads first (distinct registers)
        v2f afn = *(const v2f*)aptr;
        v2f bfn[4];
#pragma unroll
        for (int t = 0; t < 4; ++t) {
            bfn[t].x = bptr[t * 16];
            bfn[t].y = bptr[t * 16 + Nc];
        }
        // consume current fragments
#pragma unroll
        for (int t = 0; t < 4; ++t)
            acc[t] = __builtin_amdgcn_wmma_f32_16x16x4_f32(
                false, afr, false, bf[t], (short)0, acc[t], false, false);
        afr = afn;
#pragma unroll
        for (int t = 0; t < 4; ++t) bf[t] = bfn[t];
    }
    // epilogue: last iteration's WMMAs
#pragma unroll
    for (int t = 0; t < 4; ++t)
        acc[t] = __builtin_amdgcn_wmma_f32_16x16x4_f32(
            false, afr, false, bf[t], (short)0, acc[t], false, false);
}

template <int AMODE>
__global__ __launch_bounds__(64) void wmma_gemm_f32(
    const float* __restrict__ A, const float* __restrict__ kf,
    const float* __restrict__ ms, const float* __restrict__ Bm,
    const float* __restrict__ bias, const float* __restrict__ extra,
    float* __restrict__ D, int Nc, int K, int C) {
    int lane = threadIdx.x;          // 0..31
    int half = lane >> 4;            // 0/1
    int r    = lane & 15;
    int m0 = (blockIdx.y * 2 + threadIdx.y) * 16;
    int n0 = blockIdx.x * 64;
    int m  = m0 + r;                 // this lane's A row

    v8f acc[4];
#pragma unroll
    for (int t = 0; t < 4; ++t) acc[t] = (v8f){0, 0, 0, 0, 0, 0, 0, 0};

    if (AMODE == 0) {
        gemm_inner(A + (size_t)m * K, Bm, Nc, n0, r, half, 0, K, 0, acc);
    } else if (AMODE == 1) {
        gemm_inner(kf + (size_t)m * C, Bm, Nc, n0, r, half, 0, C, 0, acc);
        gemm_inner(ms + (size_t)(m % NKPT) * C, Bm, Nc, n0, r, half,
                   C, 2 * C, C, acc);
    } else {
        int mm = m >> 1;             // (m & 1) == (r & 1): uniform per lane
        const float* arow = (m & 1) ? (ms + (size_t)(mm % NKPT) * C)
                                    : (kf + (size_t)mm * C);
        gemm_inner(arow, Bm, Nc, n0, r, half, 0, K, 0, acc);
    }

    // C/D layout: VGPR v, lane -> M = (lane/16)*8 + v, N = lane%16
#pragma unroll
    for (int t = 0; t < 4; ++t) {
#pragma unroll
        for (int vv = 0; vv < 8; ++vv) {
            int mr = m0 + half * 8 + vv;
            int nc = n0 + t * 16 + r;
            float val = acc[t][vv] + bias[nc];
            if (extra) val += extra[(size_t)mr * Nc + nc];
            D[(size_t)mr * Nc + nc] = val;
        }
    }
}

// ---------------------------------------------------------------------------
// 3) Row LayerNorm (population variance, eps=1e-5) + ReLU.  One block per row.
// ---------------------------------------------------------------------------
__global__ __launch_bounds__(256) void ln_relu_kernel(
    const float* __restrict__ X, const float* __restrict__ g,
    const float* __restrict__ beta, float* __restrict__ Y, int C) {
    __shared__ float sh1[256], sh2[256];
    int m = blockIdx.x;
    const float* row = X + (size_t)m * C;
    float s = 0.0f, s2 = 0.0f;
    for (int c = threadIdx.x; c < C; c += blockDim.x) {
        float x = row[c];
        s += x; s2 += x * x;
    }
    sh1[threadIdx.x] = s; sh2[threadIdx.x] = s2;
    __syncthreads();
    for (int off = 128; off > 0; off >>= 1) {
        if ((int)threadIdx.x < off) {
            sh1[threadIdx.x] += sh1[threadIdx.x + off];
            sh2[threadIdx.x] += sh2[threadIdx.x + off];
        }
        __syncthreads();
    }
    float mean = sh1[0] / (float)C;
    float var  = sh2[0] / (float)C - mean * mean;
    float rs = rsqrtf(var + 1e-5f);
    for (int c = threadIdx.x; c < C; c += blockDim.x) {
        float x = (row[c] - mean) * rs * g[c] + beta[c];
        Y[(size_t)m * C + c] = fmaxf(x, 0.0f);
    }
}

// ---------------------------------------------------------------------------
// 4) S=2 gated multi-head attention, pre-averaged over sequence:
//    OGm[m,c] = 0.5 * sum_s gate(s,h(c)) * (softmax(qk/sqrt(d)) @ v)[s,c]
//    One block per m (8 waves = 8 heads); wave32 butterfly reductions.
// ---------------------------------------------------------------------------
__global__ __launch_bounds__(256) void attn_kernel(
    const float* __restrict__ kf, const float* __restrict__ ms,
    const float* __restrict__ QKV, const float* __restrict__ wg,
    const float* __restrict__ bg, float* __restrict__ OGm, int C) {
    int m = blockIdx.x;
    int h = threadIdx.x >> 5;        // head = wave id
    int lane = threadIdx.x & 31;
    int d = C >> 3;

    // headwise sigmoid gate from the raw sequence rows
    float ga[2];
#pragma unroll
    for (int s = 0; s < 2; ++s) {
        float acc = 0.0f;
        const float* xrow = (s == 0) ? (kf + (size_t)m * C)
                                     : (ms + (size_t)(m % NKPT) * C);
        for (int c = lane; c < C; c += 32)
            acc += xrow[c] * wg[(size_t)c * NHEADS + h];
        acc = wave_allreduce(acc);
        ga[s] = 1.0f / (1.0f + expf(-(acc + bg[h])));
    }

    const float* q0 = QKV + (size_t)(2 * m)     * (3 * C) + h * d;
    const float* q1 = QKV + (size_t)(2 * m + 1) * (3 * C) + h * d;
    const float* k0 = q0 + C;     const float* k1 = q1 + C;
    const float* v0 = q0 + 2 * C; const float* v1 = q1 + 2 * C;

    float e00 = 0, e01 = 0, e10 = 0, e11 = 0;
    for (int j = lane; j < d; j += 32) {
        float qa = q0[j], qb = q1[j], ka = k0[j], kb = k1[j];
        e00 += qa * ka; e01 += qa * kb; e10 += qb * ka; e11 += qb * kb;
    }
    e00 = wave_allreduce(e00); e01 = wave_allreduce(e01);
    e10 = wave_allreduce(e10); e11 = wave_allreduce(e11);
    float sc = rsqrtf((float)d);
    e00 *= sc; e01 *= sc; e10 *= sc; e11 *= sc;

    float mx0 = fmaxf(e00, e01), mx1 = fmaxf(e10, e11);
    float p00 = expf(e00 - mx0), p01 = expf(e01 - mx0);
    float p10 = expf(e10 - mx1), p11 = expf(e11 - mx1);
    float i0 = 1.0f / (p00 + p01), i1 = 1.0f / (p10 + p11);
    float a00 = p00 * i0, a01 = p01 * i0;
    float a10 = p10 * i1, a11 = p11 * i1;

    for (int j = lane; j < d; j += 32) {
        float o0 = a00 * v0[j] + a01 * v1[j];
        float o1 = a10 * v0[j] + a11 * v1[j];
        OGm[(size_t)m * C + h * d + j] = 0.5f * (o0 * ga[0] + o1 * ga[1]);
    }
}

// ---------------------------------------------------------------------------
// Host launcher
// ---------------------------------------------------------------------------
extern "C" void kernel_launch(void* const* d_in, const int* in_sizes, int n_in,
                              void* d_out, int out_size, void* d_ws, size_t ws_size,
                              hipStream_t stream) {
    (void)in_sizes; (void)n_in; (void)out_size; (void)ws_size;

    const float* fms[2]  = {(const float*)d_in[0], (const float*)d_in[1]};
    const float* kpts    = (const float*)d_in[2];
    const int Hs[2] = {96, 48}, Ws[2] = {72, 36}, Cs[2] = {256, 512};

    // Workspace layout (floats), sized at C=512, reused across both levels.
    float* ws   = (float*)d_ws;
    float* kf   = ws;                 // 544 * 512
    float* Hpre = ws + 278528;        // 544 * 512
    float* Hr   = ws + 557056;        // 544 * 512
    float* P    = ws + 835584;        // 544 * 512
    float* QKV  = ws + 1114112;       // 1088 * 1536
    float* OGm  = ws + 2785280;       // 544 * 512

    const int M = BATCH * NKPT;       // 544
    size_t outoff = 0;

    for (int lvl = 0; lvl < 2; ++lvl) {
        int base = 3 + lvl * 13;
        const float* ms   = (const float*)d_in[base + 0];
        const float* w1   = (const float*)d_in[base + 1];
        const float* b1   = (const float*)d_in[base + 2];
        const float* g    = (const float*)d_in[base + 3];
        const float* beta = (const float*)d_in[base + 4];
        const float* w2   = (const float*)d_in[base + 5];
        const float* b2   = (const float*)d_in[base + 6];
        const float* wqkv = (const float*)d_in[base + 7];
        const float* bqkv = (const float*)d_in[base + 8];
        const float* wg   = (const float*)d_in[base + 9];
        const float* bg   = (const float*)d_in[base + 10];
        const float* wo   = (const float*)d_in[base + 11];
        const float* bo   = (const float*)d_in[base + 12];
        int C = Cs[lvl], H = Hs[lvl], W = Ws[lvl];

        // kf = gaussian keypoint pooling
        gauss_pool_kernel<<<M, 256, 0, stream>>>(fms[lvl], kpts, kf, C, H, W);

        dim3 blk(32, 2);
        // GEMM1: Hpre = concat(kf|meta) @ w1 + b1      (M x C, K = 2C)
        dim3 g1(C / 64, M / 32);
        wmma_gemm_f32<1><<<g1, blk, 0, stream>>>(nullptr, kf, ms, w1, b1,
                                                 nullptr, Hpre, C, 2 * C, C);
        // Hr = relu(LN(Hpre))
        ln_relu_kernel<<<M, 256, 0, stream>>>(Hpre, g, beta, Hr, C);
        // P = Hr @ w2 + b2
        wmma_gemm_f32<0><<<g1, blk, 0, stream>>>(Hr, nullptr, nullptr, w2, b2,
                                                 nullptr, P, C, C, C);
        // QKV = stacked @ wqkv + bqkv                  (2M x 3C, K = C)
        dim3 g3(3 * C / 64, 2 * M / 32);
        wmma_gemm_f32<2><<<g3, blk, 0, stream>>>(nullptr, kf, ms, wqkv, bqkv,
                                                 nullptr, QKV, 3 * C, C, C);
        // OGm = seq-mean of gated attention output
        attn_kernel<<<M, 256, 0, stream>>>(kf, ms, QKV, wg, bg, OGm, C);
        // out = OGm @ wo + bo + P  (directly into d_out)
        wmma_gemm_f32<0><<<g1, blk, 0, stream>>>(OGm, nullptr, nullptr, wo, bo,
                                                 P, (float*)d_out + outoff,
                                                 C, C, C);
        outoff += (size_t)M * C;
    }
}